// GAT_27839978012782
// MI455X (gfx1250) — compile-verified
//
#include <hip/hip_runtime.h>
#include <hip/hip_bf16.h>
#include <stdint.h>

#define HEADS 8
#define HDIM  16
#define HID   128
#define NEG_SLOPE 0.2f
#define BN_EPS 1e-5f
#define SM_EPS 1e-16f

typedef __attribute__((ext_vector_type(2))) float v2f;
typedef __attribute__((ext_vector_type(8))) float v8f;

// ---------------------------------------------------------------------------
// float atomic max via sign-split int atomics (lowers to native global atomics)
// ---------------------------------------------------------------------------
__device__ __forceinline__ void atomicMaxFloat(float* addr, float value) {
  if (value >= 0.0f) {
    atomicMax((int*)addr, __float_as_int(value));
  } else {
    atomicMin((unsigned int*)addr, (unsigned int)__float_as_int(value));
  }
}

// ---------------------------------------------------------------------------
// xh = H (N x 128) @ W (128 x 128), fp32 WMMA 16x16x4.
// Block = 256 threads = 8 waves. Wave w computes the 16x16 tile at
// rows [blockIdx.x*16, +16), cols [w*16, +16). K loop: 32 x v_wmma_f32_16x16x4_f32.
//
// A layout (ISA 7.12.2, 32-bit A 16x4): lane L holds row m = L&15;
//   VGPR0/1 = A[m][2*(L>>4) + 0/1]  -> one v2f per lane.
// B layout (4x16, rows striped across lanes): lane L holds col n = L&15;
//   VGPR0/1 = B[2*(L>>4) + 0/1][n]  -> one v2f per lane.
// C/D layout: VGPR r = D[r + 8*(L>>4)][L&15].
//
// Out-of-range rows are CLAMPED (not predicated): loads stay unconditional,
// EXEC stays all-1s through the whole WMMA chain (required by ISA 7.12), and
// no per-iteration exec-save/branch is emitted. Store side masks the tail.
// ---------------------------------------------------------------------------
__global__ void gat_gemm_wmma(const float* __restrict__ H,
                              const float* __restrict__ W,
                              float* __restrict__ XH, int N) {
  const int wave  = threadIdx.x >> 5;     // 0..7 -> column tile
  const int lane  = threadIdx.x & 31;
  const int m     = lane & 15;
  const int khalf = lane >> 4;            // 0 or 1
  int row = blockIdx.x * 16 + m;
  if (row > N - 1) row = N - 1;           // clamp: keeps loads in range, no divergence
  const int col = wave * 16 + m;          // for B/D, (lane&15) is the column

  const float* __restrict__ Arow = H + (size_t)row * HID + 2 * khalf;
  const float* __restrict__ Bcol = W + (size_t)(2 * khalf) * HID + col;

  v8f c = {};
  #pragma unroll
  for (int k0 = 0; k0 < HID; k0 += 4) {
    const v2f a = *(const v2f*)(Arow + k0);
    v2f b;
    b.x = Bcol[(size_t)k0 * HID];
    b.y = Bcol[(size_t)(k0 + 1) * HID];
    c = __builtin_amdgcn_wmma_f32_16x16x4_f32(
        /*neg_a=*/false, a, /*neg_b=*/false, b,
        /*c_mod=*/(short)0, c, /*reuse_a=*/false, /*reuse_b=*/false);
  }

  #pragma unroll
  for (int r = 0; r < 8; ++r) {
    const int orow = blockIdx.x * 16 + r + 8 * khalf;
    if (orow < N) XH[(size_t)orow * HID + col] = c[r];
  }
}

// ---------------------------------------------------------------------------
// Per-node prep: attention half-logits, init segment max / denom, zero accum.
// One block (128 threads) per node.
// ---------------------------------------------------------------------------
__global__ void gat_node_prep(const float* __restrict__ xh,
                              const float* __restrict__ a_src,
                              const float* __restrict__ a_dst,
                              float* __restrict__ al_src,
                              float* __restrict__ al_dst,
                              float* __restrict__ mbuf,
                              float* __restrict__ dbuf,
                              float* __restrict__ accum, int N) {
  const int n = blockIdx.x;
  const int t = threadIdx.x;
  if (n >= N) return;
  accum[(size_t)n * HID + t] = 0.0f;
  if (t < HEADS) {
    const float* row = xh + (size_t)n * HID + t * HDIM;
    float s1 = 0.0f, s2 = 0.0f;
    #pragma unroll
    for (int d = 0; d < HDIM; ++d) {
      const float v = row[d];
      s1 += v * a_src[t * HDIM + d];
      s2 += v * a_dst[t * HDIM + d];
    }
    al_src[n * HEADS + t] = s1;
    al_dst[n * HEADS + t] = s2;
    mbuf[n * HEADS + t] = -__builtin_inff();
    dbuf[n * HEADS + t] = 0.0f;
  }
}

__device__ __forceinline__ void edge_ids(int e, const int* __restrict__ src,
                                         const int* __restrict__ dst, int E,
                                         int& s, int& d) {
  if (e < E) { s = src[e]; d = dst[e]; }
  else       { s = e - E;  d = e - E;  }   // self loops
}

// pass 1: segment max of leaky-relu logits
__global__ void gat_edge_max(const int* __restrict__ src, const int* __restrict__ dst,
                             const float* __restrict__ als, const float* __restrict__ ald,
                             float* __restrict__ mbuf, int E, int N) {
  const int total = (E + N) * HEADS;
  for (int i = blockIdx.x * blockDim.x + threadIdx.x; i < total;
       i += gridDim.x * blockDim.x) {
    const int h = i & (HEADS - 1);
    const int e = i >> 3;
    int s, d; edge_ids(e, src, dst, E, s, d);
    float v = als[s * HEADS + h] + ald[d * HEADS + h];
    v = (v > 0.0f) ? v : NEG_SLOPE * v;
    atomicMaxFloat(mbuf + d * HEADS + h, v);
  }
}

// pass 2: denom = segment sum of exp(e - m)
__global__ void gat_edge_sum(const int* __restrict__ src, const int* __restrict__ dst,
                             const float* __restrict__ als, const float* __restrict__ ald,
                             const float* __restrict__ mbuf, float* __restrict__ dbuf,
                             int E, int N) {
  const int total = (E + N) * HEADS;
  for (int i = blockIdx.x * blockDim.x + threadIdx.x; i < total;
       i += gridDim.x * blockDim.x) {
    const int h = i & (HEADS - 1);
    const int e = i >> 3;
    int s, d; edge_ids(e, src, dst, E, s, d);
    float v = als[s * HEADS + h] + ald[d * HEADS + h];
    v = (v > 0.0f) ? v : NEG_SLOPE * v;
    const float ex = expf(v - mbuf[d * HEADS + h]);
    atomicAdd(dbuf + d * HEADS + h, ex);
  }
}

// pass 3: accum[dst] += alpha * xh[src]   (128 f32 atomics per edge)
__global__ void gat_edge_scatter(const int* __restrict__ src, const int* __restrict__ dst,
                                 const float* __restrict__ als, const float* __restrict__ ald,
                                 const float* __restrict__ mbuf, const float* __restrict__ dbuf,
                                 const float* __restrict__ xh, float* __restrict__ accum,
                                 int E, int N) {
  const int total = (E + N) * HEADS;
  for (int i = blockIdx.x * blockDim.x + threadIdx.x; i < total;
       i += gridDim.x * blockDim.x) {
    const int h = i & (HEADS - 1);
    const int e = i >> 3;
    int s, d; edge_ids(e, src, dst, E, s, d);
    float v = als[s * HEADS + h] + ald[d * HEADS + h];
    v = (v > 0.0f) ? v : NEG_SLOPE * v;
    const float ex = expf(v - mbuf[d * HEADS + h]);
    const float alpha = ex / (dbuf[d * HEADS + h] + SM_EPS);
    const float4* xr = (const float4*)(xh + (size_t)s * HID + h * HDIM);
    float* out = accum + (size_t)d * HID + h * HDIM;
    #pragma unroll
    for (int q = 0; q < 4; ++q) {
      const float4 vv = xr[q];
      atomicAdd(out + 4 * q + 0, vv.x * alpha);
      atomicAdd(out + 4 * q + 1, vv.y * alpha);
      atomicAdd(out + 4 * q + 2, vv.z * alpha);
      atomicAdd(out + 4 * q + 3, vv.w * alpha);
    }
  }
}

// bias + BatchNorm(eval) + ReLU + residual
__global__ void gat_post(const float* __restrict__ hin, const float* __restrict__ accum,
                         const float* __restrict__ bias, const float* __restrict__ gamma,
                         const float* __restrict__ beta, const float* __restrict__ mean,
                         const float* __restrict__ var, float* __restrict__ hout, int N) {
  const int idx = blockIdx.x * blockDim.x + threadIdx.x;
  if (idx >= N * HID) return;
  const int c = idx & (HID - 1);
  float v = accum[idx] + bias[c];
  v = gamma[c] * (v - mean[c]) * rsqrtf(var[c] + BN_EPS) + beta[c];
  v = (v > 0.0f) ? v : 0.0f;
  hout[idx] = hin[idx] + v;
}

// y[n] = h[n,:] . out_W + out_b    (block of 128 threads per node)
__global__ void gat_final(const float* __restrict__ h, const float* __restrict__ W,
                          const float* __restrict__ b, float* __restrict__ out, int N) {
  __shared__ float red[4];
  const int n = blockIdx.x;
  const int t = threadIdx.x;
  if (n >= N) return;
  float v = h[(size_t)n * HID + t] * W[t];
  #pragma unroll
  for (int off = 16; off > 0; off >>= 1) v += __shfl_down(v, off);
  if ((t & 31) == 0) red[t >> 5] = v;
  __syncthreads();
  if (t == 0) out[n] = red[0] + red[1] + red[2] + red[3] + b[0];
}

// ---------------------------------------------------------------------------
extern "C" void kernel_launch(void* const* d_in, const int* in_sizes, int n_in,
                              void* d_out, int out_size, void* d_ws, size_t ws_size,
                              hipStream_t stream) {
  const float* x       = (const float*)d_in[0];
  const int*   ei      = (const int*)d_in[1];
  const float* Ws      = (const float*)d_in[2];
  const float* att_src = (const float*)d_in[3];
  const float* att_dst = (const float*)d_in[4];
  const float* biases  = (const float*)d_in[5];
  const float* gamma   = (const float*)d_in[6];
  const float* beta    = (const float*)d_in[7];
  const float* bnmean  = (const float*)d_in[8];
  const float* bnvar   = (const float*)d_in[9];
  const float* outW    = (const float*)d_in[10];
  const float* outb    = (const float*)d_in[11];

  const int N = in_sizes[0] / HID;
  const int E = in_sizes[1] / 2;
  const int L = in_sizes[2] / (HID * HID);
  const int* src = ei;
  const int* dst = ei + E;

  // carve workspace
  char* ws = (char*)d_ws;
  size_t off = 0;
  auto carve = [&](size_t bytes) -> void* {
    void* p = ws + off;
    off += (bytes + 255) & ~(size_t)255;
    return p;
  };
  float* xh    = (float*)carve((size_t)N * HID * sizeof(float));
  float* accum = (float*)carve((size_t)N * HID * sizeof(float));
  float* h0    = (float*)carve((size_t)N * HID * sizeof(float));
  float* h1    = (float*)carve((size_t)N * HID * sizeof(float));
  float* alsrc = (float*)carve((size_t)N * HEADS * sizeof(float));
  float* aldst = (float*)carve((size_t)N * HEADS * sizeof(float));
  float* mbuf  = (float*)carve((size_t)N * HEADS * sizeof(float));
  float* dbuf  = (float*)carve((size_t)N * HEADS * sizeof(float));
  (void)ws_size; (void)n_in; (void)out_size;

  const int edgeBlocks = 4096, edgeThreads = 256;
  const float* hin = x;
  float* hbufs[2] = {h0, h1};

  for (int l = 0; l < L; ++l) {
    gat_gemm_wmma<<<(N + 15) / 16, 256, 0, stream>>>(hin, Ws + (size_t)l * HID * HID, xh, N);
    gat_node_prep<<<N, HID, 0, stream>>>(xh, att_src + (size_t)l * HID, att_dst + (size_t)l * HID,
                                         alsrc, aldst, mbuf, dbuf, accum, N);
    gat_edge_max<<<edgeBlocks, edgeThreads, 0, stream>>>(src, dst, alsrc, aldst, mbuf, E, N);
    gat_edge_sum<<<edgeBlocks, edgeThreads, 0, stream>>>(src, dst, alsrc, aldst, mbuf, dbuf, E, N);
    gat_edge_scatter<<<edgeBlocks, edgeThreads, 0, stream>>>(src, dst, alsrc, aldst, mbuf, dbuf,
                                                             xh, accum, E, N);
    float* hout = hbufs[l & 1];
    gat_post<<<((size_t)N * HID + 255) / 256, 256, 0, stream>>>(
        hin, accum, biases + (size_t)l * HID, gamma + (size_t)l * HID, beta + (size_t)l * HID,
        bnmean + (size_t)l * HID, bnvar + (size_t)l * HID, hout, N);
    hin = hout;
  }

  gat_final<<<N, HID, 0, stream>>>(hin, outW, outb, (float*)d_out, N);
}